// GCN_10161892623141
// MI455X (gfx1250) — compile-verified
//
#include <hip/hip_runtime.h>

typedef __attribute__((ext_vector_type(2))) float v2f;
typedef __attribute__((ext_vector_type(8))) float v8f;

// ---------------- utility kernels ----------------

__global__ void fill_kernel(float* __restrict__ p, float v, int n) {
    int i = blockIdx.x * blockDim.x + threadIdx.x;
    if (i < n) p[i] = v;
}

__global__ void degree_kernel(const int* __restrict__ dst, float* __restrict__ deg, int E) {
    int e = blockIdx.x * blockDim.x + threadIdx.x;
    if (e < E) atomicAdd(&deg[dst[e]], 1.0f);
}

__global__ void rsqrt_kernel(float* __restrict__ p, int n) {
    int i = blockIdx.x * blockDim.x + threadIdx.x;
    if (i < n) p[i] = rsqrtf(p[i]);
}

// ---------------- fp32 WMMA GEMM: H[N,OUTD] = X[N,IND] * W[IND,OUTD] ----------------
// One wave computes a 16-row strip across all OUTD/16 column tiles using
// V_WMMA_F32_16X16X4_F32 (exact fp32, K=4 per issue).
//
// ISA layouts (wave32):
//   A 16x4 f32 : lane&15 = M, float2 = { K=2*(lane>>4), K=2*(lane>>4)+1 }
//   B 4x16 f32 : lane&15 = N, float2 = { B[2h][n], B[2h+1][n] }
//   C/D 16x16  : 8 VGPRs; lanes 0-15 -> M=v, lanes 16-31 -> M=v+8, N=lane&15
template <int IND, int OUTD>
__global__ void gemm_wmma_kernel(const float* __restrict__ X,
                                 const float* __restrict__ W,
                                 float* __restrict__ H, int nRows) {
    const int wave = threadIdx.x >> 5;
    const int lane = threadIdx.x & 31;
    const int row  = lane & 15;
    const int half = lane >> 4;
    const int m0   = (blockIdx.x * (blockDim.x >> 5) + wave) << 4;
    if (m0 >= nRows) return;  // wave-uniform; grid sized for exact coverage

    constexpr int CT = OUTD / 16;
    v8f acc[CT] = {};

    const float* __restrict__ xrow = X + (size_t)(m0 + row) * IND + 2 * half;
    for (int k = 0; k < IND; k += 4) {
        v2f a;
        a.x = xrow[k];
        a.y = xrow[k + 1];
        const float* __restrict__ w0 = W + (size_t)(k + 2 * half) * OUTD + row;
        const float* __restrict__ w1 = w0 + OUTD;
#pragma unroll
        for (int ct = 0; ct < CT; ++ct) {
            v2f b;
            b.x = w0[ct * 16];
            b.y = w1[ct * 16];
            acc[ct] = __builtin_amdgcn_wmma_f32_16x16x4_f32(
                false, a, false, b, (short)0, acc[ct], false, false);
        }
    }
#pragma unroll
    for (int ct = 0; ct < CT; ++ct)
#pragma unroll
        for (int v = 0; v < 8; ++v)
            H[(size_t)(m0 + v + 8 * half) * OUTD + ct * 16 + row] = acc[ct][v];
}

// ---------------- self-loop + bias init: AGG = H * invs^2 + b ----------------
__global__ void init_agg_kernel(const float* __restrict__ H, const float* __restrict__ invs,
                                const float* __restrict__ bias, float* __restrict__ AGG,
                                int n, int D) {
    size_t i = (size_t)blockIdx.x * blockDim.x + threadIdx.x;
    if (i >= (size_t)n * D) return;
    int r = (int)(i / D);
    int c = (int)(i - (size_t)r * D);
    float s = invs[r];
    AGG[i] = H[i] * s * s + bias[c];
}

// ---------------- edge scatter: AGG[dst] += H[src] * invs[src]*invs[dst] ----------------
// D/4 lanes per edge, float4 per lane -> one wave handles one edge when D==128,
// giving perfectly coalesced 512B row reads (rows live in the 192MB L2).
__global__ void scatter_kernel(const float* __restrict__ H, const int* __restrict__ src,
                               const int* __restrict__ dst, const float* __restrict__ invs,
                               float* __restrict__ AGG, int E, int D) {
    const int perEdge = D >> 2;
    long long t = (long long)blockIdx.x * blockDim.x + threadIdx.x;
    int e = (int)(t / perEdge);
    if (e >= E) return;
    int c = ((int)(t % perEdge)) << 2;
    int s = src[e], d = dst[e];
    float coef = invs[s] * invs[d];
    const float4 hv = *(const float4*)(H + (size_t)s * D + c);
    float* op = AGG + (size_t)d * D + c;
    atomicAdd(op + 0, hv.x * coef);
    atomicAdd(op + 1, hv.y * coef);
    atomicAdd(op + 2, hv.z * coef);
    atomicAdd(op + 3, hv.w * coef);
}

// ---------------- batchnorm: column sums (D fixed at 128) ----------------
__global__ void bn_stats_kernel(const float* __restrict__ A, float* __restrict__ colsum,
                                float* __restrict__ colsq, int n) {
    int c  = threadIdx.x;               // 128 threads = 128 columns
    int r0 = blockIdx.x * 256;
    int r1 = r0 + 256 < n ? r0 + 256 : n;
    float s = 0.f, sq = 0.f;
    for (int r = r0; r < r1; ++r) {
        float v = A[(size_t)r * 128 + c];
        s += v;
        sq += v * v;
    }
    atomicAdd(&colsum[c], s);
    atomicAdd(&colsq[c], sq);
}

__global__ void bn_relu_kernel(float* __restrict__ A, const float* __restrict__ colsum,
                               const float* __restrict__ colsq, const float* __restrict__ g,
                               const float* __restrict__ bt, int n) {
    size_t i = (size_t)blockIdx.x * blockDim.x + threadIdx.x;
    if (i >= (size_t)n * 128) return;
    int c = (int)(i & 127);
    float invN = 1.0f / (float)n;
    float mean = colsum[c] * invN;
    float var  = colsq[c] * invN - mean * mean;
    float v = g[c] * (A[i] - mean) * rsqrtf(var + 1e-5f) + bt[c];
    A[i] = v > 0.f ? v : 0.f;
}

// ---------------- driver ----------------

extern "C" void kernel_launch(void* const* d_in, const int* in_sizes, int n_in,
                              void* d_out, int out_size, void* d_ws, size_t ws_size,
                              hipStream_t stream) {
    const float* x   = (const float*)d_in[0];
    const int*   ei  = (const int*)d_in[1];
    const float* W1  = (const float*)d_in[2];
    const float* b1  = (const float*)d_in[3];
    const float* g1  = (const float*)d_in[4];
    const float* bt1 = (const float*)d_in[5];
    const float* W2  = (const float*)d_in[6];
    const float* b2  = (const float*)d_in[7];
    const float* g2  = (const float*)d_in[8];
    const float* bt2 = (const float*)d_in[9];
    const float* W3  = (const float*)d_in[10];
    const float* b3  = (const float*)d_in[11];
    float* out = (float*)d_out;

    const int N  = in_sizes[0] / 128;   // 50000
    const int E  = in_sizes[1] / 2;     // 800000
    const int D3 = in_sizes[10] / 128;  // 64
    const int* src = ei;
    const int* dst = ei + E;

    // workspace layout
    float* bufH   = (float*)d_ws;              // N*128 (GEMM output h)
    float* bufA   = bufH + (size_t)N * 128;    // N*128 (agg / activations)
    float* invs   = bufA + (size_t)N * 128;    // N
    float* colsum = invs + N;                  // 128
    float* colsq  = colsum + 128;              // 128

    const int T = 256;
    auto blocks = [](long long n, int t) { return (unsigned)((n + t - 1) / t); };

    // ---- degrees -> inv_sqrt (graph is shared across layers) ----
    fill_kernel<<<blocks(N, T), T, 0, stream>>>(invs, 1.0f, N);
    degree_kernel<<<blocks(E, T), T, 0, stream>>>(dst, invs, E);
    rsqrt_kernel<<<blocks(N, T), T, 0, stream>>>(invs, N);

    const int rowTiles  = N / 16;              // 3125
    const unsigned gGemm = (unsigned)(rowTiles / 5);  // 625 blocks x 5 waves: exact cover

    // ================= layer 1 =================
    gemm_wmma_kernel<128, 128><<<gGemm, 160, 0, stream>>>(x, W1, bufH, N);
    init_agg_kernel<<<blocks((long long)N * 128, T), T, 0, stream>>>(bufH, invs, b1, bufA, N, 128);
    scatter_kernel<<<blocks((long long)E * 32, T), T, 0, stream>>>(bufH, src, dst, invs, bufA, E, 128);
    fill_kernel<<<1, 256, 0, stream>>>(colsum, 0.f, 256);  // zeros colsum+colsq
    bn_stats_kernel<<<blocks(N, 256), 128, 0, stream>>>(bufA, colsum, colsq, N);
    bn_relu_kernel<<<blocks((long long)N * 128, T), T, 0, stream>>>(bufA, colsum, colsq, g1, bt1, N);

    // ================= layer 2 =================
    gemm_wmma_kernel<128, 128><<<gGemm, 160, 0, stream>>>(bufA, W2, bufH, N);
    init_agg_kernel<<<blocks((long long)N * 128, T), T, 0, stream>>>(bufH, invs, b2, bufA, N, 128);
    scatter_kernel<<<blocks((long long)E * 32, T), T, 0, stream>>>(bufH, src, dst, invs, bufA, E, 128);
    fill_kernel<<<1, 256, 0, stream>>>(colsum, 0.f, 256);
    bn_stats_kernel<<<blocks(N, 256), 128, 0, stream>>>(bufA, colsum, colsq, N);
    bn_relu_kernel<<<blocks((long long)N * 128, T), T, 0, stream>>>(bufA, colsum, colsq, g2, bt2, N);

    // ================= layer 3 (no BN, straight to d_out) =================
    gemm_wmma_kernel<128, 64><<<gGemm, 160, 0, stream>>>(bufA, W3, bufH, N);
    init_agg_kernel<<<blocks((long long)N * D3, T), T, 0, stream>>>(bufH, invs, b3, out, N, D3);
    scatter_kernel<<<blocks((long long)E * (D3 / 4), T), T, 0, stream>>>(bufH, src, dst, invs, out, E, D3);
}